// bicon_loss_46600395162340
// MI455X (gfx1250) — compile-verified
//
#include <hip/hip_runtime.h>
#include <hip/hip_bf16.h>

typedef __attribute__((ext_vector_type(2))) float v2f;
typedef __attribute__((ext_vector_type(8))) float v8f;

#define B_    16
#define C_    8
#define H_    512
#define W_    512
#define SEG   32
#define CHUNK (H_ / SEG)          // 16 rows per thread
#define NCOL  (B_ * W_)           // 8192 dice columns
#define N1F   ((float)(B_ * H_ * W_))
#define N8F   (8.0f * N1F)

// Wave(32)-wide sum using V_WMMA_F32_16X16X4_F32 with only documented layouts:
//   A (16x4, fully documented): lane L<16 holds A(L,0)=v, A(L,1)=0;
//                               lane L>=16 holds A(L-16,2)=v, A(L-16,3)=0.
//   B = all-ones (layout-invariant), C = 0.
//   => D(m,n) = v[m] + v[m+16] for every n.
//   D (16x16, documented): lane L<16, VGPR r = D(r,L); lane L>=16, VGPR r = D(r+8,L-16).
//   Sum of a lane's 8 D VGPRs = half-sum; shfl_xor(16) adds the other half.
// Requires EXEC all-ones (call only from converged full-wave code).
__device__ __forceinline__ float wave_sum_wmma(float v) {
    v2f a; a[0] = v;    a[1] = 0.0f;
    v2f b; b[0] = 1.0f; b[1] = 1.0f;
    v8f c = {};
    v8f d = __builtin_amdgcn_wmma_f32_16x16x4_f32(
        /*neg_a=*/false, a, /*neg_b=*/false, b,
        /*c_mod=*/(short)0, c, /*reuse_a=*/false, /*reuse_b=*/false);
    float s = d[0] + d[1] + d[2] + d[3] + d[4] + d[5] + d[6] + d[7];
    s += __shfl_xor(s, 16, 32);
    return s;
}

__global__ void bicon_init_kernel(float* __restrict__ ws, int n) {
    int i = blockIdx.x * blockDim.x + threadIdx.x;
    if (i < n) ws[i] = 0.0f;
}

// DIRS[i] = (dx, dy); vote_i(h,w) = p_i(h,w) * p_{7-i}(h-dy, w-dx), zero-filled OOB.
__global__ void __launch_bounds__(256) bicon_main_kernel(
    const float* __restrict__ cmap,     // [B,8,H,W]
    const int*   __restrict__ target,   // [B,1,H,W]
    const int*   __restrict__ con,      // [B,8,H,W]
    float*       __restrict__ ws)       // [0]=scalar acc, then 3*NCOL col sums
{
    const int tid  = threadIdx.x;
    const int wb   = blockIdx.x & 1;            // which 256-wide half of W
    const int rest = blockIdx.x >> 1;
    const int seg  = rest & (SEG - 1);
    const int b    = rest >> 5;                 // SEG == 32
    const int w    = wb * 256 + tid;
    const int h0   = seg * CHUNK;

    const int DXa[8] = { 1, 0,-1, 1,-1, 1, 0,-1};
    const int DYa[8] = { 1, 1, 1, 0, 0,-1,-1,-1};

    const float* cm_b = cmap + (size_t)b * C_ * H_ * W_;
    const int*   cn_b = con  + (size_t)b * C_ * H_ * W_;
    const int*   tg_b = target + (size_t)b * H_ * W_;

    float conmap_l = 0.0f, bimap_l = 0.0f, dec_l = 0.0f, bce_l = 0.0f;
    float tcol = 0.0f, fcol = 0.0f, icol = 0.0f;

    for (int h = h0; h < h0 + CHUNK; ++h) {
        float p[8], lnp[8], ln1mp[8];
        int   ct[8];
        #pragma unroll
        for (int c = 0; c < 8; ++c) {
            size_t idx = ((size_t)c * H_ + h) * W_ + w;
            float x  = cm_b[idx];
            ct[c]    = cn_b[idx];
            float ex = __expf(-x);
            float t1 = 1.0f + ex;
            float lt = __logf(t1);
            p[c]     = __builtin_amdgcn_rcpf(t1);   // sigmoid(x) via v_rcp_f32
            lnp[c]   = -lt;                          // log p
            ln1mp[c] = -x - lt;                      // log(1-p)
        }
        const int t = tg_b[h * W_ + w];

        float csum = 0.0f, bsum = 0.0f;
        float maxv = 0.0f, maxlv = -1e30f, maxl1v = -1e30f;
        int   sconn = 0;
        #pragma unroll
        for (int i = 0; i < 8; ++i) {
            const int nh = h - DYa[i];
            const int nw = w - DXa[i];
            const int nc = 7 - i;
            const bool valid = (nh >= 0) && (nh < H_) && (nw >= 0) && (nw < W_);
            float q = 0.0f, lnq = -1e30f;
            if (valid) {
                float xn  = cm_b[((size_t)nc * H_ + nh) * W_ + nw];
                float exn = __expf(-xn);
                float tn  = 1.0f + exn;
                q   = __builtin_amdgcn_rcpf(tn);    // sigmoid via v_rcp_f32
                lnq = -__logf(tn);
            }
            const float v   = p[i] * q;             // vote_out
            const float lv  = lnp[i] + lnq;         // log(vote)
            const float l1v = __logf(1.0f - v);     // log(1-vote)
            const int cti = ct[i];
            sconn += cti;
            csum += -fmaxf(cti ? lnp[i] : ln1mp[i], -100.0f);  // conmap BCE
            bsum += -fmaxf(cti ? lv     : l1v,      -100.0f);  // bimap BCE
            maxv   = fmaxf(maxv,   v);
            maxlv  = fmaxf(maxlv,  lv);    // == log(max v)
            maxl1v = fmaxf(maxl1v, l1v);   // == log(1 - min v)
        }
        conmap_l += csum;
        bimap_l  += bsum;

        // BCE(final_pred, target)
        const float l1fp = __logf(1.0f - maxv);
        bce_l += -fmaxf(t ? maxlv : l1fp, -100.0f);

        // decouple: edge pixels only; pred_min = min(v)*edge; log(1-pred_min)=maxl1v
        const bool edge = (sconn > 0) && (sconn < 8);
        dec_l += edge ? -fmaxf(maxl1v, -100.0f) : 0.0f;

        // dice column partials
        tcol += (float)t;
        fcol += maxv;
        icol += t ? maxv : 0.0f;
    }

    // Linear terms collapse into one pre-weighted scalar per thread.
    float part = (0.8f / N8F) * conmap_l + (0.2f / N8F) * bimap_l
               + (1.0f / N1F) * (dec_l + bce_l);

    // Converged here: WMMA-based wave reduction (EXEC all-ones).
    float wsum = wave_sum_wmma(part);

    __shared__ float red[8];
    const int wave = tid >> 5, lane = tid & 31;
    if (lane == 0) red[wave] = wsum;
    __syncthreads();
    if (tid == 0) {
        float s = 0.0f;
        #pragma unroll
        for (int i = 0; i < 8; ++i) s += red[i];
        atomicAdd(&ws[0], s);
    }

    const int col = b * W_ + w;
    atomicAdd(&ws[1 + col],            tcol);
    atomicAdd(&ws[1 + NCOL + col],     fcol);
    atomicAdd(&ws[1 + 2 * NCOL + col], icol);
}

__global__ void __launch_bounds__(1024) bicon_final_kernel(
    const float* __restrict__ ws, float* __restrict__ out)
{
    const float* tarr = ws + 1;
    const float* farr = ws + 1 + NCOL;
    const float* iarr = ws + 1 + 2 * NCOL;
    float d = 0.0f;
    #pragma unroll
    for (int col = threadIdx.x; col < NCOL; col += 1024) {
        const float ts = tarr[col], fs = farr[col], is = iarr[col];
        d += 1.0f - (2.0f * is + 0.001f) / (ts + fs + 0.001f);
    }
    float wsum = wave_sum_wmma(d);   // converged, EXEC all-ones

    __shared__ float red[32];
    const int wave = threadIdx.x >> 5, lane = threadIdx.x & 31;
    if (lane == 0) red[wave] = wsum;
    __syncthreads();
    if (threadIdx.x == 0) {
        float s = 0.0f;
        #pragma unroll
        for (int i = 0; i < 32; ++i) s += red[i];
        out[0] = ws[0] + s / (float)NCOL;   // + loss_dice
    }
}

extern "C" void kernel_launch(void* const* d_in, const int* in_sizes, int n_in,
                              void* d_out, int out_size, void* d_ws, size_t ws_size,
                              hipStream_t stream) {
    const float* cmap   = (const float*)d_in[0];
    const int*   target = (const int*)d_in[1];
    const int*   con    = (const int*)d_in[2];
    float* ws  = (float*)d_ws;
    float* out = (float*)d_out;

    const int nws = 1 + 3 * NCOL;  // ~98 KB of scratch
    hipLaunchKernelGGL(bicon_init_kernel, dim3((nws + 255) / 256), dim3(256), 0, stream,
                       ws, nws);
    hipLaunchKernelGGL(bicon_main_kernel, dim3(2 * B_ * SEG), dim3(256), 0, stream,
                       cmap, target, con, ws);
    hipLaunchKernelGGL(bicon_final_kernel, dim3(1), dim3(1024), 0, stream, ws, out);
}